// AdditiveAttention_24902220383034
// MI455X (gfx1250) — compile-verified
//
#include <hip/hip_runtime.h>
#include <hip/hip_bf16.h>
#include <math.h>

// ---------------------------------------------------------------------------
// Bahdanau additive attention, fused for MI455X (gfx1250, wave32).
//   B=32, S=2048, H=1024.
//   Dominant work: v_proj GEMM [65536x1024]x[1024x1024] (137 GFLOP fp32)
//   via V_WMMA_F32_16X16X4_F32 (exact fp32 on the matrix pipe), fused with
//   tanh(q_proj + v_proj) and the .Wc reduction so the 256MB feat/v_proj
//   intermediates never touch HBM.
//   Wv is repacked once into WMMA B-fragment lane order (2 K-steps per
//   float4) so each global_load_b128 feeds two WMMAs from L2.
//   values tile staged into LDS with gfx1250 async-to-LDS DMA.
// ---------------------------------------------------------------------------

typedef __attribute__((ext_vector_type(2))) float v2f;
typedef __attribute__((ext_vector_type(8))) float v8f;

#define BB 32       // batch
#define SS 2048     // sequence
#define HH 1024     // hidden
#define MTILE 32    // values rows per workgroup in score kernel
#define LDSROW 1028 // padded LDS row stride (floats): conflict-free A loads

#if __has_builtin(__builtin_amdgcn_wmma_f32_16x16x4_f32)
#define USE_WMMA_F32X4 1
#else
#define USE_WMMA_F32X4 0   // host pass fallback
#endif

#if defined(__HIP_DEVICE_COMPILE__)
#define USE_ASYNC_STAGE 1  // gfx1250 global_load_async_to_lds_b128 path
#else
#define USE_ASYNC_STAGE 0
#endif

// ---------------------------------------------------------------------------
// Kernel 0: repack Wv [k][n] into B-fragment order:
//   pw[((nt*128 + kp)*32 + lane)*4 + c] with
//     half = lane>>4, lm = lane&15, ks = 2*kp + (c>>1)
//     value = Wv[(4*ks + half*2 + (c&1)) * H + nt*16 + lm]
// One float4 per lane per kp covers K-steps {8kp, 8kp+4} for its WMMA pair.
// 4MB total, L2-resident, repacked once per launch (negligible).
// ---------------------------------------------------------------------------
__global__ __launch_bounds__(256) void packWv_kernel(
    const float* __restrict__ Wv, float* __restrict__ pw) {
  const int gid  = blockIdx.x * 256 + threadIdx.x;  // 0 .. 2^20-1
  const int c    = gid & 3;
  const int lane = (gid >> 2) & 31;
  const int kp   = (gid >> 7) & 127;
  const int nt   = gid >> 14;
  const int half = lane >> 4;
  const int lm   = lane & 15;
  const int ks   = 2 * kp + (c >> 1);
  const int krow = 4 * ks + (half << 1) + (c & 1);
  pw[gid] = Wv[krow * HH + nt * 16 + lm];
}

// ---------------------------------------------------------------------------
// Kernel 1: wsq[b][n] = (query @ Wq)[b][n] + bq[n] + bv[n]   (tiny)
// ---------------------------------------------------------------------------
__global__ __launch_bounds__(256) void qproj_kernel(
    const float* __restrict__ query, const float* __restrict__ Wq,
    const float* __restrict__ bq, const float* __restrict__ bv,
    float* __restrict__ wsq) {
  const int b = blockIdx.x;
  const int t = threadIdx.x;
  float a0 = 0.f, a1 = 0.f, a2 = 0.f, a3 = 0.f;
  for (int k = 0; k < HH; ++k) {
    const float qv = query[b * HH + k];  // uniform -> scalar load
    const float* w = Wq + k * HH + t;
    a0 += qv * w[0];
    a1 += qv * w[256];
    a2 += qv * w[512];
    a3 += qv * w[768];
  }
  wsq[b * HH + t      ] = a0 + bq[t      ] + bv[t      ];
  wsq[b * HH + t + 256] = a1 + bq[t + 256] + bv[t + 256];
  wsq[b * HH + t + 512] = a2 + bq[t + 512] + bv[t + 512];
  wsq[b * HH + t + 768] = a3 + bq[t + 768] + bv[t + 768];
}

// ---------------------------------------------------------------------------
// Kernel 2 (dominant): 32 values-rows per workgroup, 8 waves.
//   Async-DMA 32x1024 fp32 values tile into padded LDS.
//   Wave w: M-subtile (w>>2)*16, N-tiles {(w&3)+4j}. Per N-tile: 128 kp-steps,
//   each = 1 global_load_b128 (packed Wv) + 1 ds_load_2addr_b64 (A) feeding
//   2x v_wmma_f32_16x16x4_f32. Epilogue: tanh + Wc reduction via ds_add_f32.
// ---------------------------------------------------------------------------
__global__ __launch_bounds__(256) void score_kernel(
    const float* __restrict__ values, const float* __restrict__ pw,
    const float* __restrict__ Wc, const float* __restrict__ wsq,
    float* __restrict__ wscores) {
  extern __shared__ float smem[];
  float* ldsA = smem;                   // MTILE x LDSROW floats
  float* sc   = smem + MTILE * LDSROW;  // MTILE partial scores

  const int t    = threadIdx.x;
  const int row0 = blockIdx.x * MTILE;  // never crosses a batch boundary
  const int b    = row0 >> 11;

  if (t < MTILE) sc[t] = 0.f;

  // ---- stage contiguous 32x1024 fp32 tile (128KB) into padded LDS --------
  const float* srcf = values + (size_t)row0 * HH;
#if USE_ASYNC_STAGE
#pragma unroll
  for (int it = 0; it < 32; ++it) {
    const int g4 = t + it * 256;  // float4 index
    const int g  = g4 << 2;
    const int r  = g >> 10;
    const int c  = g & (HH - 1);
    unsigned ldsOff = (unsigned)(size_t)&ldsA[r * LDSROW + c]; // LDS byte addr (low 32b)
    unsigned long long ga = (unsigned long long)(size_t)(srcf + g);
    asm volatile("global_load_async_to_lds_b128 %0, %1, off"
                 :: "v"(ldsOff), "v"(ga) : "memory");
  }
  asm volatile("s_wait_asynccnt 0x0" ::: "memory");
#else
  const float4* src4 = (const float4*)srcf;
#pragma unroll
  for (int it = 0; it < 32; ++it) {
    const int g4 = t + it * 256;
    const int g  = g4 << 2;
    const int r  = g >> 10;
    const int c  = g & (HH - 1);
    *(float4*)&ldsA[r * LDSROW + c] = src4[g4];
  }
#endif
  __syncthreads();

  const int lane   = t & 31;
  const int wv     = t >> 5;
  const int half   = lane >> 4;
  const int lm     = lane & 15;
  const int msbase = (wv >> 2) << 4;  // 0 or 16

  // A fragment base: row = msbase+lm, K-pair offset half*2 within each k-step
  const float* ap = &ldsA[(msbase + lm) * LDSROW + (half << 1)];

  for (int j = 0; j < 16; ++j) {
    const int nt = (wv & 3) + (j << 2);  // N-tile 0..63
    const int n  = (nt << 4) + lm;

    // Packed B strip: one float4 per kp per lane, coalesced across the wave.
    const float4* bp = (const float4*)pw + ((size_t)nt * 128) * 32 + lane;
    __builtin_prefetch(bp, 0, 1);  // global_prefetch_b8

    v8f acc = {};
#pragma unroll 4
    for (int kp = 0; kp < 128; ++kp) {
      const int k = kp << 3;                 // covers K-steps k and k+4
      v2f a0 = *(const v2f*)(ap + k);        // ds_load_2addr_b64 (merged)
      v2f a1 = *(const v2f*)(ap + k + 4);
      const float4 b4 = bp[kp * 32];         // global_load_b128 from L2
      v2f b0; b0.x = b4.x; b0.y = b4.y;
      v2f b1; b1.x = b4.z; b1.y = b4.w;
#if USE_WMMA_F32X4
      acc = __builtin_amdgcn_wmma_f32_16x16x4_f32(
          false, a0, false, b0, (short)0, acc, false, false);
      acc = __builtin_amdgcn_wmma_f32_16x16x4_f32(
          false, a1, false, b1, (short)0, acc, false, false);
#else
      acc[0] += a0.x * b0.x + a0.y * b0.y;   // host-pass placeholder
      acc[1] += a1.x * b1.x + a1.y * b1.y;
#endif
    }

    // feat = tanh(acc + (q_proj+bq+bv)[b][n]); score += feat * Wc[n]
    const float qpb = wsq[b * HH + n];
    const float wcn = Wc[n];
#pragma unroll
    for (int i = 0; i < 8; ++i) {
      const float tt = tanhf(acc[i] + qpb) * wcn;
      atomicAdd(&sc[msbase + i + (half << 3)], tt);  // ds_add_f32
    }
  }
  __syncthreads();
  if (t < MTILE) wscores[row0 + t] = sc[t];
}

// ---------------------------------------------------------------------------
// Kernel 3: masked softmax over S per batch.
// ---------------------------------------------------------------------------
__global__ __launch_bounds__(256) void softmax_kernel(
    const float* __restrict__ wscores, const unsigned char* __restrict__ mask,
    const float* __restrict__ bc, float* __restrict__ wout) {
  const int b = blockIdx.x;
  const int t = threadIdx.x;
  __shared__ float red[256];
  const float bc0 = bc[0];
  const float NEG = -3.402823466e38f;

  float v[8];
  float lmax = NEG;
#pragma unroll
  for (int j = 0; j < 8; ++j) {
    const int s = t + j * 256;
    float x = wscores[b * SS + s] + bc0;
    if (!mask[b * SS + s]) x = NEG;
    v[j] = x;
    lmax = fmaxf(lmax, x);
  }
  red[t] = lmax;
  __syncthreads();
  for (int o = 128; o > 0; o >>= 1) {
    if (t < o) red[t] = fmaxf(red[t], red[t + o]);
    __syncthreads();
  }
  const float gmax = red[0];
  __syncthreads();

  float lsum = 0.f;
#pragma unroll
  for (int j = 0; j < 8; ++j) {
    v[j] = expf(v[j] - gmax);
    lsum += v[j];
  }
  red[t] = lsum;
  __syncthreads();
  for (int o = 128; o > 0; o >>= 1) {
    if (t < o) red[t] += red[t + o];
    __syncthreads();
  }
  const float inv = 1.0f / red[0];
#pragma unroll
  for (int j = 0; j < 8; ++j) wout[b * SS + t + j * 256] = v[j] * inv;
}

// ---------------------------------------------------------------------------
// Kernel 4: context[b][h] = sum_s weights[b][s] * values[b][s][h]
// ---------------------------------------------------------------------------
__global__ __launch_bounds__(256) void context_kernel(
    const float* __restrict__ values, const float* __restrict__ wout,
    float* __restrict__ ctx) {
  const int b = blockIdx.x;
  const int t = threadIdx.x;  // owns 4 consecutive h
  const float* w = wout + b * SS;
  const float4* vp = (const float4*)(values + (size_t)b * SS * HH);
  float4 acc = make_float4(0.f, 0.f, 0.f, 0.f);
#pragma unroll 4
  for (int s = 0; s < SS; ++s) {
    const float ws = w[s];                     // uniform -> scalar
    const float4 x = vp[(size_t)s * 256 + t];  // coalesced b128
    acc.x += ws * x.x;
    acc.y += ws * x.y;
    acc.z += ws * x.z;
    acc.w += ws * x.w;
  }
  ((float4*)(ctx + b * HH))[t] = acc;
}

// ---------------------------------------------------------------------------
extern "C" void kernel_launch(void* const* d_in, const int* in_sizes, int n_in,
                              void* d_out, int out_size, void* d_ws, size_t ws_size,
                              hipStream_t stream) {
  const float* query  = (const float*)d_in[0];
  const float* values = (const float*)d_in[1];
  const unsigned char* mask = (const unsigned char*)d_in[2];
  const float* Wq = (const float*)d_in[3];
  const float* bq = (const float*)d_in[4];
  const float* Wv = (const float*)d_in[5];
  const float* bv = (const float*)d_in[6];
  const float* Wc = (const float*)d_in[7];
  const float* bc = (const float*)d_in[8];

  float* out     = (float*)d_out;
  float* ctx     = out;            // [32,1024]
  float* weights = out + BB * HH;  // [32,2048]

  float* wsq     = (float*)d_ws;        // 32768 floats
  float* wscores = wsq + BB * HH;       // 65536 floats
  float* pw      = wscores + BB * SS;   // 1048576 floats (4MB packed Wv)

  packWv_kernel<<<(HH * HH) / 256, 256, 0, stream>>>(Wv, pw);
  qproj_kernel<<<BB, 256, 0, stream>>>(query, Wq, bq, bv, wsq);

  const size_t smem = (size_t)(MTILE * LDSROW + MTILE) * sizeof(float);  // ~128.6KB
  score_kernel<<<(BB * SS) / MTILE, 256, smem, stream>>>(values, pw, Wc, wsq, wscores);

  softmax_kernel<<<BB, 256, 0, stream>>>(wscores, mask, bc, weights);

  context_kernel<<<BB, 256, 0, stream>>>(values, weights, ctx);
}